// CostVolume_66529043415439
// MI455X (gfx1250) — compile-verified
//
#include <hip/hip_runtime.h>
#include <hip/hip_bf16.h>

// Problem constants (from the reference)
#define BATCH 4
#define CHANS 320
#define HEIGHT 64
#define WIDTH 128
#define GROUPS 40
#define CG 8            // channels per group
#define DISP 48         // disparity planes
#define HW (HEIGHT * WIDTH)      // 8192
#define LSTRIDE 144     // LDS row stride in floats (padded: 2*144 % 64 == 32 -> no
                        // bank overlap between lane halves reading k and k+2 rows)

typedef __attribute__((ext_vector_type(2))) float v2f;
typedef __attribute__((ext_vector_type(4))) float v4f;   // native clang vector (NT-store ok)
typedef __attribute__((ext_vector_type(8))) float v8f;

__global__ __launch_bounds__(256)
void gwc_costvolume_wmma_kernel(const float* __restrict__ left,
                                const float* __restrict__ right,
                                float* __restrict__ out) {
    // One block per (b, g, h)
    const int bgh = blockIdx.x;
    const int h = bgh & (HEIGHT - 1);
    const int g = (bgh / HEIGHT) % GROUPS;
    const int b = bgh / (HEIGHT * GROUPS);

    const int tid  = threadIdx.x;
    const int lane = tid & 31;
    // Wave index as a *scalar* so tile selection is an s_cmp/s_cbranch,
    // not a v_cmp + exec-mask dance around the WMMAs.
    const int wave = __builtin_amdgcn_readfirstlane(tid >> 5);

    __shared__ float sL[CG * LSTRIDE];          // left  features, [c][w], padded rows
    __shared__ float sR[CG * LSTRIDE];          // right features, [c][w], padded rows
    __shared__ float slab[DISP * WIDTH];        // output slab [d][w] for this (b,g,h)
    __shared__ float dumpw[32];                 // per-lane sink for out-of-band lanes

    // ---- Stage L/R rows into LDS: 8 channels x 128 floats = 256 v4f each ----
    {
        const size_t base = ((size_t)(b * CHANS + g * CG) * HEIGHT + h) * WIDTH;
        const v4f* __restrict__ gl = reinterpret_cast<const v4f*>(left + base);
        const v4f* __restrict__ gr = reinterpret_cast<const v4f*>(right + base);
        const int c  = tid >> 5;        // 0..7
        const int w4 = tid & 31;        // 0..31 (v4f index within row)
        const int gidx = c * (HW / 4) + w4;
        reinterpret_cast<v4f*>(&sL[c * LSTRIDE])[w4] = gl[gidx];
        reinterpret_cast<v4f*>(&sR[c * LSTRIDE])[w4] = gr[gidx];
    }

    // ---- Zero the slab (masked band entries must read back as 0) ----
    {
        v4f z = {0.f, 0.f, 0.f, 0.f};
        v4f* s4 = reinterpret_cast<v4f*>(slab);
        #pragma unroll
        for (int it = 0; it < 6; ++it)             // 6*256 = 1536 v4f = 48*128 floats
            s4[tid + it * 256] = z;
    }

    __syncthreads();

    // ---- Gram-band tiles via V_WMMA_F32_16X16X4_F32 (M = w', N = w, K = channel) ----
    // For each row tile wi, column tiles wj = wi - {0,16,32,48} (clamped >= 0) cover
    // the full band d = w - w' in [0,48). Each (w,w') pair lands in exactly one tile.
    const int m  = lane & 15;   // A: M index (w' within tile) AND B/D: N index (w within tile)
    const int kh = lane >> 4;   // lane half selects K pair {0,1} vs {2,3}

    int t = 0;
    for (int i = 0; i < 8; ++i) {
        const int wi = i * 16;
        const int nd = (i < 3) ? (i + 1) : 4;
        for (int j = 0; j < nd; ++j, ++t) {
            if ((t & 7) != wave) continue;          // scalar: s_cmp + s_cbranch
            const int wj = wi - j * 16;

            // A = R tile (16x4, M=w', K=c), B = L tile (4x16, K=c, N=w)
            v2f a, bv;
            a.x  = sR[(2 * kh + 0) * LSTRIDE + wj + m];
            a.y  = sR[(2 * kh + 1) * LSTRIDE + wj + m];
            bv.x = sL[(2 * kh + 0) * LSTRIDE + wi + m];
            bv.y = sL[(2 * kh + 1) * LSTRIDE + wi + m];

            v8f acc = {};
            // (neg_a, A, neg_b, B, c_mod, C, reuse_a, reuse_b)
            acc = __builtin_amdgcn_wmma_f32_16x16x4_f32(
                false, a, false, bv, (short)0, acc, false, false);

            // K = 4..7
            a.x  = sR[(4 + 2 * kh + 0) * LSTRIDE + wj + m];
            a.y  = sR[(4 + 2 * kh + 1) * LSTRIDE + wj + m];
            bv.x = sL[(4 + 2 * kh + 0) * LSTRIDE + wi + m];
            bv.y = sL[(4 + 2 * kh + 1) * LSTRIDE + wi + m];
            acc = __builtin_amdgcn_wmma_f32_16x16x4_f32(
                false, a, false, bv, (short)0, acc, false, false);

            // Scatter into slab: D layout -> VGPR r, lane: M = r + 8*kh, N = lane%16.
            // Branch-free band mask: out-of-band lanes write a per-lane dump word
            // (address select -> v_cndmask + unconditional ds_store, no exec churn).
            const int w = wi + m;                   // left column (N)
            #pragma unroll
            for (int r = 0; r < 8; ++r) {
                const int wp = wj + r + 8 * kh;     // right column (M)
                const unsigned du = (unsigned)(w - wp);   // d, unsigned-range check
                float* p = (du < DISP) ? (slab + (du * WIDTH + w)) : (dumpw + lane);
                *p = acc[r] * 0.125f;               // mean over Cg=8
            }
        }
    }

    __syncthreads();

    // ---- Stream slab to global: coalesced v4f non-temporal stores ----
    {
        float* outp = out + ((size_t)(b * GROUPS + g) * DISP) * HW + (size_t)h * WIDTH;
        const v4f* s4 = reinterpret_cast<const v4f*>(slab);
        #pragma unroll
        for (int it = 0; it < 6; ++it) {
            const int idx = tid + it * 256;         // 0..1535
            const int d   = idx >> 5;               // plane
            const int w4  = idx & 31;               // v4f chunk within plane
            v4f v = s4[idx];                        // slab is [d][w] contiguous
            __builtin_nontemporal_store(
                v, reinterpret_cast<v4f*>(outp + (size_t)d * HW) + w4);
        }
    }
}

extern "C" void kernel_launch(void* const* d_in, const int* in_sizes, int n_in,
                              void* d_out, int out_size, void* d_ws, size_t ws_size,
                              hipStream_t stream) {
    (void)in_sizes; (void)n_in; (void)d_ws; (void)ws_size; (void)out_size;
    const float* left  = (const float*)d_in[0];   // left_gwc  [4,320,64,128]
    const float* right = (const float*)d_in[1];   // right_gwc [4,320,64,128]
    // d_in[2], d_in[3] (unary features) are unused by the 'gwc' reference path.
    float* out = (float*)d_out;                   // [4,40,48,64,128]

    const int nblocks = BATCH * GROUPS * HEIGHT;  // 10240 blocks, one per (b,g,h)
    gwc_costvolume_wmma_kernel<<<nblocks, 256, 0, stream>>>(left, right, out);
}